// ContrastiveDistortion_65910568125121
// MI455X (gfx1250) — compile-verified
//
#include <hip/hip_runtime.h>
#include <math.h>

typedef __attribute__((ext_vector_type(2))) float v2f;
typedef __attribute__((ext_vector_type(8))) float v8f;
typedef int v4i __attribute__((vector_size(16)));   // matches builtin param type

#define N_TOT   4096
#define BATCH   2048
#define DDIM    128
#define KDIM    512            // concat feature length (2 * 2 * D)
#define TILE    16
#define NTILES  (N_TOT / TILE) // 256
#define ROWS_WG 64
#define WAVES   4
#define LDS_PITCH (KDIM + 4)   // pad 4 dwords -> conflict-free b64 lane access

#if defined(__has_builtin)
#  if __has_builtin(__builtin_amdgcn_global_load_async_to_lds_b128)
#    define HAVE_ASYNC_LDS 1
#  endif
#endif
#ifndef HAVE_ASYNC_LDS
#  define HAVE_ASYNC_LDS 0
#endif

#if HAVE_ASYNC_LDS
typedef __attribute__((address_space(1))) v4i* gasync_ptr;
typedef __attribute__((address_space(3))) v4i* lasync_ptr;
#endif

// -------------------------------------------------------------------------
// Kernel 1: build fused features F[n][0:512] and r[n] = 0.25 * sum(mu^2*inv)
//   F = [ 0.5*inv | -(mu*inv) | var+mu^2 | mu ]
//   G (B-side of the GEMM) is a block-swap of F -> F[col][k ^ 256]
// -------------------------------------------------------------------------
__global__ __launch_bounds__(DDIM)
void prep_kernel(const float* __restrict__ mu_x, const float* __restrict__ sigma_x,
                 const float* __restrict__ mu_p, const float* __restrict__ sigma_p,
                 float* __restrict__ F, float* __restrict__ rvec) {
    const int row = blockIdx.x;     // 0..4095
    const int d   = threadIdx.x;    // 0..127

    const float* mu_src = (row < BATCH) ? (mu_x    + (size_t)row * DDIM)
                                        : (mu_p    + (size_t)(row - BATCH) * DDIM);
    const float* sg_src = (row < BATCH) ? (sigma_x + (size_t)row * DDIM)
                                        : (sigma_p + (size_t)(row - BATCH) * DDIM);
    const float mu  = mu_src[d];
    const float sg  = sg_src[d];
    const float var = sg * sg;
    const float inv = 1.0f / var;
    const float w   = var + mu * mu;
    const float mi  = mu * inv;

    float* Fr = F + (size_t)row * KDIM;
    Fr[d]            =  0.5f * inv;
    Fr[DDIM + d]     = -mi;
    Fr[2 * DDIM + d] =  w;
    Fr[3 * DDIM + d] =  mu;

    __shared__ float sred[DDIM];
    sred[d] = mu * mi;              // mu^2 * inv
    __syncthreads();
    for (int off = DDIM / 2; off > 0; off >>= 1) {
        if (d < off) sred[d] += sred[d + off];
        __syncthreads();
    }
    if (d == 0) rvec[row] = 0.25f * sred[0];
}

// -------------------------------------------------------------------------
// Kernel 2: fused symmetric-KL GEMM (WMMA f32 16x16x4) + online log-softmax.
// One WG = 64 rows (132KB LDS, async-staged), 4 waves x one 16-row strip.
// -------------------------------------------------------------------------
__global__ __launch_bounds__(WAVES * 32)
void kl_softmax_kernel(const float* __restrict__ F, const float* __restrict__ rvec,
                       float* __restrict__ partials) {
    __shared__ float ldsF[ROWS_WG * LDS_PITCH];   // 132096 B
    __shared__ float red_m[ROWS_WG][16];
    __shared__ float red_s[ROWS_WG][16];
    __shared__ float pos_l[ROWS_WG];
    __shared__ float terms[ROWS_WG];

    const int tid  = threadIdx.x;
    const int wg   = blockIdx.x;
    const int row0 = wg * ROWS_WG;

    // ---- stage this WG's 64 F-rows into LDS (async DMA if available) ----
    const float4* Fsrc = (const float4*)(F + (size_t)row0 * KDIM);
#if HAVE_ASYNC_LDS
    for (int i = tid; i < ROWS_WG * (KDIM / 4); i += WAVES * 32) {
        const int r = i / (KDIM / 4);
        const int c = i % (KDIM / 4);
        __builtin_amdgcn_global_load_async_to_lds_b128(
            (gasync_ptr)(Fsrc + (size_t)r * (KDIM / 4) + c),
            (lasync_ptr)&ldsF[r * LDS_PITCH + c * 4],
            0, 0);
    }
    asm volatile("s_wait_asynccnt 0x0" ::: "memory");
#else
    for (int i = tid; i < ROWS_WG * (KDIM / 4); i += WAVES * 32) {
        const int r = i / (KDIM / 4);
        const int c = i % (KDIM / 4);
        float4 v = Fsrc[(size_t)r * (KDIM / 4) + c];
        *(float4*)&ldsF[r * LDS_PITCH + c * 4] = v;
    }
#endif
    __syncthreads();

    const int wave  = tid >> 5;
    const int lane  = tid & 31;
    const int ln16  = lane & 15;
    const int hi    = lane >> 4;       // 0: K+{0,1}, 1: K+{2,3} (ISA 32-bit A/B layout)
    const int rloc  = wave * TILE;
    const int rglob = row0 + rloc;

    const float invT    = 10.0f;                                 // 1 / TEMPERATURE
    const float to_add  = -logf((4095.0f - 0.2f) / 4094.0f);     // weight=5 positive bump
    const float NEG_BIG = -3.0e38f;                              // finite -inf sentinel

    float run_m[8], run_s[8];
#pragma unroll
    for (int j = 0; j < 8; ++j) { run_m[j] = NEG_BIG; run_s[j] = 0.0f; }

    // A-fragment base in LDS for this lane (row = rloc + ln16, k offset 2*hi)
    const float* aBase = &ldsF[(rloc + ln16) * LDS_PITCH + (hi << 1)];

    for (int t = 0; t < NTILES; ++t) {
        const int colg = t * TILE + ln16;
        // G[col][k] = F[col][k ^ 256]; split K into two halves so both A and B
        // use constant VGPR bases + immediate offsets (no per-load address ALU)
        const float* bLo = F + (size_t)colg * KDIM + (hi << 1);  // serves k0 in [256,512)
        const float* bHi = bLo + 256;                            // serves k0 in [0,256)
        if (t + 1 < NTILES)
            __builtin_prefetch(bLo + (size_t)TILE * KDIM, 0, 0);

        v8f acc = {};
#pragma unroll 8
        for (int k0 = 0; k0 < 256; k0 += 4) {
            v2f a = *(const v2f*)(aBase + k0);
            v2f b = *(const v2f*)(bHi + k0);
            acc = __builtin_amdgcn_wmma_f32_16x16x4_f32(
                false, a, false, b, (short)0, acc, false, false);
        }
#pragma unroll 8
        for (int k0 = 0; k0 < 256; k0 += 4) {
            v2f a = *(const v2f*)(aBase + 256 + k0);
            v2f b = *(const v2f*)(bLo + k0);
            acc = __builtin_amdgcn_wmma_f32_16x16x4_f32(
                false, a, false, b, (short)0, acc, false, false);
        }

        const float rb = rvec[colg];
#pragma unroll
        for (int j = 0; j < 8; ++j) {
            const int rowg = rglob + j + (hi << 3);   // C layout: VGPR j -> M=j (+8 hi half)
            float l = fmaf(acc[j], -0.5f, -rb) * invT;
            const int pos = (rowg < BATCH) ? rowg + BATCH : rowg - BATCH;
            const bool isPos = (colg == pos);
            if (isPos) pos_l[rloc + j + (hi << 3)] = l + to_add;  // one store per row total
            l = isPos ? l + to_add : l;
            l = (colg == rowg) ? NEG_BIG : l;         // drop diagonal, branchless
            const float m  = run_m[j];
            const float nm = fmaxf(m, l);
            run_s[j] = run_s[j] * __expf(m - nm) + __expf(l - nm);
            run_m[j] = nm;
        }
    }

    // ---- cross-lane LSE merge: 16 column-classes per row ----
#pragma unroll
    for (int j = 0; j < 8; ++j) {
        const int rl = rloc + j + (hi << 3);
        red_m[rl][ln16] = run_m[j];
        red_s[rl][ln16] = run_s[j];
    }
    __syncthreads();

    if (tid < ROWS_WG) {
        float M = NEG_BIG;
        for (int i = 0; i < 16; ++i) M = fmaxf(M, red_m[tid][i]);
        float S = 0.0f;
        for (int i = 0; i < 16; ++i) S += red_s[tid][i] * __expf(red_m[tid][i] - M);
        const float lse = M + __logf(S);
        terms[tid] = pos_l[tid] - lse;                // logp at positive
    }
    __syncthreads();

    if (tid == 0) {
        float s = 0.0f;
        for (int i = 0; i < ROWS_WG; ++i) s += terms[i];
        partials[wg] = s;                             // deterministic (no atomics)
    }
}

// -------------------------------------------------------------------------
// Kernel 3: final scalar reduction
// -------------------------------------------------------------------------
__global__ void finalize_kernel(const float* __restrict__ partials, float* __restrict__ out) {
    if (threadIdx.x == 0 && blockIdx.x == 0) {
        float s = 0.0f;
        for (int i = 0; i < N_TOT / ROWS_WG; ++i) s += partials[i];
        out[0] = -s / (float)N_TOT;
    }
}

extern "C" void kernel_launch(void* const* d_in, const int* in_sizes, int n_in,
                              void* d_out, int out_size, void* d_ws, size_t ws_size,
                              hipStream_t stream) {
    // inputs: z_hat(unused), mu_x, sigma_x, mu_p, sigma_p  (all float32)
    const float* mu_x    = (const float*)d_in[1];
    const float* sigma_x = (const float*)d_in[2];
    const float* mu_p    = (const float*)d_in[3];
    const float* sigma_p = (const float*)d_in[4];
    float* out = (float*)d_out;

    // workspace layout
    float* F        = (float*)d_ws;                            // 4096*512 f32 = 8 MB
    float* rvec     = F + (size_t)N_TOT * KDIM;                // 4096 f32
    float* partials = rvec + N_TOT;                            // 64 f32

    prep_kernel<<<N_TOT, DDIM, 0, stream>>>(mu_x, sigma_x, mu_p, sigma_p, F, rvec);
    kl_softmax_kernel<<<N_TOT / ROWS_WG, WAVES * 32, 0, stream>>>(F, rvec, partials);
    finalize_kernel<<<1, 64, 0, stream>>>(partials, out);
}